// GlobalSelfAttention_28544352649554
// MI455X (gfx1250) — compile-verified
//
#include <hip/hip_runtime.h>
#include <hip/hip_bf16.h>
#include <math.h>
#include <stdint.h>

typedef __attribute__((ext_vector_type(2))) float v2f;
typedef __attribute__((ext_vector_type(8))) float v8f;

// Problem constants (B, S, D from the reference; T == S, self-attention)
constexpr int Bc = 2;
constexpr int Tc = 400;
constexpr int Sc = 400;
constexpr int Dc = 512;
constexpr int TT = 8;    // query rows (t) per block in the attention kernel
constexpr int CH = 8;    // uh rows staged in LDS per chunk
constexpr int SP = 404;  // score-row pitch: 404*l mod 64 banks is conflict-free

// ---------------------------------------------------------------------------
// Kernel 1: Y[M,N] = X[M,K] * W[K,N] (+ bias), fp32 via V_WMMA_F32_16X16X4_F32.
// One wave (32 lanes) owns a 16x16 output tile; K stepped by 4 per WMMA.
// A-frag (ISA 7.12.2, 32-bit A 16x4): lanes 0-15 -> K={0,1}, lanes 16-31 ->
// K={2,3}.  B-frag mirrored.  C/D: VGPR r = row mBase+r (+8 for upper half),
// col = lane&15.
// ---------------------------------------------------------------------------
__global__ __launch_bounds__(32) void gemm_bias_wmma(
    const float* __restrict__ X, const float* __restrict__ W,
    const float* __restrict__ bias, float* __restrict__ Y, int K) {
  const int nBase = blockIdx.x * 16;
  const int mBase = blockIdx.y * 16;
  const int lane  = threadIdx.x;
  const int half  = lane >> 4;
  const int l16   = lane & 15;

  const float* xrow = X + (size_t)(mBase + l16) * K;
  v8f acc = {};
  for (int k = 0; k < K; k += 4) {
    const int k0 = k + 2 * half;
    v2f a, b;
    a.x = xrow[k0];
    a.y = xrow[k0 + 1];
    b.x = W[(size_t)k0 * Dc + nBase + l16];
    b.y = W[(size_t)(k0 + 1) * Dc + nBase + l16];
    acc = __builtin_amdgcn_wmma_f32_16x16x4_f32(
        false, a, false, b, (short)0, acc, false, false);
  }
  const float bv = bias ? bias[nBase + l16] : 0.0f;
#pragma unroll
  for (int r = 0; r < 8; ++r) {
    const int m = mBase + r + 8 * half;
    Y[(size_t)m * Dc + nBase + l16] = acc[r] + bv;
  }
}

// ---------------------------------------------------------------------------
// Kernel 2: attention core, 8 query rows per block (one t per wave32).
//  - wq row + v live in 16 VGPRs/lane each (only ever read at e = lane+32i).
//  - uh rows double-buffered through LDS with GLOBAL_LOAD_ASYNC_TO_LDS_B128:
//    chunk k+1 issued before computing on chunk k, s_wait_asynccnt 0 +
//    barrier afterwards (ASYNCcnt semantics, async-tensor guide 4.1).
//  - scores: v . v_tanh_f32(wq + uh), wave-local shuffle reduction.
//  - per-wave masked softmax (no block barriers), align emitted [T,B,S].
//  - context P[8,400] x mem[400,512] on V_WMMA_F32_16X16X4_F32 with A
//    zero-padded to 16 rows in LDS (pitch 404 => bank-conflict-free A-frags).
// ---------------------------------------------------------------------------
__global__ __launch_bounds__(256) void attn_scores_softmax_ctx(
    const float* __restrict__ wq,   // [B*T, D]
    const float* __restrict__ uh,   // [B*S, D]
    const float* __restrict__ mem,  // [B*S, D]
    const int*   __restrict__ lens, // [B]
    const float* __restrict__ v,    // [D]
    float* __restrict__ ctx,        // [B*T, D]
    float* __restrict__ align_out)  // [T, B, S]
{
  __shared__ float uh_ch[2 * CH * Dc];  // 32 KB double-buffered key chunks
  __shared__ float sc[16 * SP];         // ~25 KB scores, zero-padded to 16 rows

  const int tilesPerB = Tc / TT;                 // 50
  const int b     = blockIdx.x / tilesPerB;
  const int tBase = (blockIdx.x % tilesPerB) * TT;
  const int tid   = threadIdx.x;
  const int lane  = tid & 31;
  const int wave  = tid >> 5;                    // 8 waves, wave == local t
  const int t     = tBase + wave;
  const int len   = lens[b];

  const float* uhB  = uh  + (size_t)b * Sc * Dc;
  const float* memB = mem + (size_t)b * Sc * Dc;

  // Phase 1: wq row + v into registers; zero the score tile's padding rows.
  float qreg[16], vreg[16];
  {
    const float* wqrow_g = wq + (size_t)(b * Tc + t) * Dc;
#pragma unroll
    for (int i = 0; i < 16; ++i) {
      qreg[i] = wqrow_g[lane + 32 * i];
      vreg[i] = v[lane + 32 * i];
    }
  }
  for (int i = tid; i < 8 * SP; i += 256) sc[8 * SP + i] = 0.0f;

  // Async-stage one CH-row chunk of uh into LDS buffer bufIdx (per-lane B128).
  auto issue_chunk = [&](int bufIdx, int s0) {
    const char* gbase = (const char*)(uhB + (size_t)s0 * Dc);
    const uint32_t lbase = (uint32_t)(uintptr_t)(&uh_ch[bufIdx * CH * Dc]);
    for (int off = tid * 16; off < CH * Dc * 4; off += 256 * 16) {
      const uint32_t l = lbase + off;
      const uint64_t g = (uint64_t)(uintptr_t)(gbase + off);
      asm volatile("global_load_async_to_lds_b128 %0, %1, off"
                   :: "v"(l), "v"(g) : "memory");
    }
  };

  issue_chunk(0, 0);
  asm volatile("s_wait_asynccnt 0x0" ::: "memory");
  __syncthreads();

  // Phase 2: scores, streaming uh through the double buffer.
  int cur = 0;
  for (int s0 = 0; s0 < Sc; s0 += CH) {
    if (s0 + CH < Sc) issue_chunk(cur ^ 1, s0 + CH);  // overlap next copy

#pragma unroll
    for (int j = 0; j < CH; ++j) {
      const int s = s0 + j;
      const float* uhrow = uh_ch + cur * CH * Dc + j * Dc;
      float acc = 0.0f;
#pragma unroll
      for (int i = 0; i < 16; ++i) {
        const float x = qreg[i] + uhrow[lane + 32 * i];
        float th;
        asm volatile("v_tanh_f32 %0, %1" : "=v"(th) : "v"(x));
        acc = fmaf(vreg[i], th, acc);
      }
#pragma unroll
      for (int off = 16; off > 0; off >>= 1) acc += __shfl_xor(acc, off, 32);
      if (lane == 0) {
        const bool valid = (s < len) && (s != t);
        sc[wave * SP + s] = valid ? acc : -INFINITY;
      }
    }
    asm volatile("s_wait_asynccnt 0x0" ::: "memory");  // next chunk landed
    __syncthreads();                                   // all waves done w/ cur
    cur ^= 1;
  }

  // Phase 3: per-wave masked softmax over this wave's 400 scores.
  float* scrow = sc + wave * SP;
  float lmax = -INFINITY;
  for (int s = lane; s < Sc; s += 32) lmax = fmaxf(lmax, scrow[s]);
#pragma unroll
  for (int off = 16; off > 0; off >>= 1)
    lmax = fmaxf(lmax, __shfl_xor(lmax, off, 32));
  float lsum = 0.0f;
  for (int s = lane; s < Sc; s += 32) {
    const float e = __expf(scrow[s] - lmax);
    scrow[s] = e;
    lsum += e;
  }
#pragma unroll
  for (int off = 16; off > 0; off >>= 1) lsum += __shfl_xor(lsum, off, 32);
  const float inv = 1.0f / lsum;
  for (int s = lane; s < Sc; s += 32) {
    const float p = scrow[s] * inv;
    scrow[s] = p;
    align_out[(size_t)(t * Bc + b) * Sc + s] = p;
  }
  __syncthreads();  // all 16 A-rows (8 probs + 8 zeros) final before WMMA

  // Phase 4: context = P[16(zero-padded),400] x mem[400,512] via WMMA.
  // Each wave owns 4 of the 32 N-tiles. Lanes 16-31 accumulate the zero rows.
  const int half = lane >> 4;
  const int l16  = lane & 15;
#pragma unroll
  for (int nt = 0; nt < 4; ++nt) {
    const int n = (wave + nt * 8) * 16 + l16;
    v8f acc = {};
    for (int k = 0; k < Sc; k += 4) {
      const int k0 = k + 2 * half;
      v2f a, bfrag;
      a.x = sc[l16 * SP + k0];
      a.y = sc[l16 * SP + k0 + 1];
      bfrag.x = memB[(size_t)k0 * Dc + n];
      bfrag.y = memB[(size_t)(k0 + 1) * Dc + n];
      acc = __builtin_amdgcn_wmma_f32_16x16x4_f32(
          false, a, false, bfrag, (short)0, acc, false, false);
    }
    if (half == 0) {  // rows 0-7 hold the 8 real contexts
#pragma unroll
      for (int r = 0; r < 8; ++r)
        ctx[(size_t)(b * Tc + tBase + r) * Dc + n] = acc[r];
    }
  }
}

// ---------------------------------------------------------------------------
// Kernel 3: attn_h = concat(ctx, input) @ Wout + bout, written [T,B,D].
// ---------------------------------------------------------------------------
__global__ __launch_bounds__(32) void out_gemm_wmma(
    const float* __restrict__ Cctx, // [B*T, D]
    const float* __restrict__ Inp,  // [B*T, D]
    const float* __restrict__ W,    // [2D, D]
    const float* __restrict__ bias, // [D]
    float* __restrict__ Y)          // [T, B, D]
{
  const int nBase = blockIdx.x * 16;
  const int mBase = blockIdx.y * 16;
  const int lane  = threadIdx.x;
  const int half  = lane >> 4;
  const int l16   = lane & 15;

  const float* crow = Cctx + (size_t)(mBase + l16) * Dc;
  const float* irow = Inp  + (size_t)(mBase + l16) * Dc;
  v8f acc = {};
  for (int k = 0; k < 2 * Dc; k += 4) {
    const int k0 = k + 2 * half;
    const float* arow = (k < Dc) ? crow : irow;
    const int kk = (k < Dc) ? k0 : (k0 - Dc);
    v2f a, b;
    a.x = arow[kk];
    a.y = arow[kk + 1];
    b.x = W[(size_t)k0 * Dc + nBase + l16];
    b.y = W[(size_t)(k0 + 1) * Dc + nBase + l16];
    acc = __builtin_amdgcn_wmma_f32_16x16x4_f32(
        false, a, false, b, (short)0, acc, false, false);
  }
  const float bv = bias[nBase + l16];
#pragma unroll
  for (int r = 0; r < 8; ++r) {
    const int m  = mBase + r + 8 * half;  // m = b*T + t; tile stays in one b
    const int bb = m / Tc;
    const int tt = m % Tc;
    Y[(size_t)(tt * Bc + bb) * Dc + nBase + l16] = acc[r] + bv;
  }
}

extern "C" void kernel_launch(void* const* d_in, const int* in_sizes, int n_in,
                              void* d_out, int out_size, void* d_ws, size_t ws_size,
                              hipStream_t stream) {
  (void)in_sizes; (void)n_in; (void)out_size; (void)ws_size;
  const float* input = (const float*)d_in[0];
  const float* memb  = (const float*)d_in[1];
  const int*   lens  = (const int*)d_in[2];
  const float* Wq    = (const float*)d_in[3];
  const float* bq    = (const float*)d_in[4];
  const float* Wc    = (const float*)d_in[5];
  const float* v     = (const float*)d_in[6];
  const float* Wout  = (const float*)d_in[7];
  const float* bout  = (const float*)d_in[8];

  float* out       = (float*)d_out;
  float* attn_out  = out;                         // [T,B,D]
  float* align_out = out + (size_t)Tc * Bc * Dc;  // [T,B,S]

  float* ws    = (float*)d_ws;
  float* wq_ws = ws;                              // [B*T, D]
  float* uh_ws = ws + (size_t)Bc * Tc * Dc;       // [B*S, D]
  float* c_ws  = ws + 2 * (size_t)Bc * Tc * Dc;   // [B*T, D]

  const dim3 gGemm(Dc / 16, (Bc * Tc) / 16);      // 32 x 50 tiles, 1 wave each

  gemm_bias_wmma<<<gGemm, 32, 0, stream>>>(input, Wq, bq, wq_ws, Dc);
  gemm_bias_wmma<<<gGemm, 32, 0, stream>>>(memb,  Wc, nullptr, uh_ws, Dc);
  attn_scores_softmax_ctx<<<Bc * (Tc / TT), 256, 0, stream>>>(
      wq_ws, uh_ws, memb, lens, v, c_ws, align_out);
  out_gemm_wmma<<<gGemm, 32, 0, stream>>>(c_ws, input, Wout, bout, attn_out);
}